// GIAttPNP_83494164234353
// MI455X (gfx1250) — compile-verified
//
#include <hip/hip_runtime.h>
#include <math.h>

#define NF       128
#define NGRAPHS  64
#define ALPHA_F  0.1f

typedef float v2f __attribute__((ext_vector_type(2)));
typedef float v8f __attribute__((ext_vector_type(8)));

// ---- monotonic float <-> unsigned encoding for atomicMax-based segment max ----
__device__ __forceinline__ unsigned flip_f32(float f) {
    unsigned u = __float_as_uint(f);
    return (u & 0x80000000u) ? ~u : (u | 0x80000000u);
}
__device__ __forceinline__ float unflip_f32(unsigned u) {
    return (u & 0x80000000u) ? __uint_as_float(u & 0x7FFFFFFFu)
                             : __uint_as_float(~u);
}

// ---- init per-graph reduction state ----
__global__ void init_seg_kernel(unsigned* __restrict__ gmax_enc, float* __restrict__ ssum) {
    int g = threadIdx.x;
    if (g < NGRAPHS) { gmax_enc[g] = 0u; ssum[g] = 0.0f; }
}

// ---- gate[n] = dot(feat[n], W[iter]) + b[iter]  via V_WMMA_F32_16X16X4_F32 ----
// One wave (32 lanes) computes a 16-node tile. B columns all hold the same W
// chunk, so every D column equals the desired dot product.
__global__ __launch_bounds__(32) void gate_wmma_kernel(
    const float* __restrict__ feat, const float* __restrict__ W,
    const float* __restrict__ b, int iter,
    float* __restrict__ gate, int n_nodes)
{
    const int base = blockIdx.x * 16;
    const int lane = threadIdx.x;          // 0..31, full wave
    const int m    = lane & 15;            // A-matrix row within tile
    const int kh   = lane >> 4;            // 0: K{0,1}, 1: K{2,3}

    int row = base + m;
    if (row >= n_nodes) row = n_nodes - 1; // clamp reads; writes are guarded
    const float* __restrict__ frow = feat + (size_t)row * NF;
    const float* __restrict__ wrow = W + (size_t)iter * NF;

    v8f c = {};
    #pragma unroll 4
    for (int k = 0; k < NF; k += 4) {
        v2f a, bm;
        // A 16x4 f32 layout: lanes 0-15 -> K=0(VGPR0),1(VGPR1); lanes 16-31 -> K=2,3
        a.x  = frow[k + 2 * kh];
        a.y  = frow[k + 2 * kh + 1];
        // B 4x16 with all 16 columns identical: per-lane value depends only on K
        bm.x = wrow[k + 2 * kh];
        bm.y = wrow[k + 2 * kh + 1];
        c = __builtin_amdgcn_wmma_f32_16x16x4_f32(
                /*neg_a=*/false, a, /*neg_b=*/false, bm,
                /*c_mod=*/(short)0, c, /*reuse_a=*/false, /*reuse_b=*/false);
    }

    const float bias = b[iter];
    // D (16x16 f32) layout: lane 0 holds rows 0..7 (col 0), lane 16 rows 8..15
    if (lane == 0) {
        #pragma unroll
        for (int j = 0; j < 8; ++j) {
            int r = base + j;
            if (r < n_nodes) gate[r] = c[j] + bias;
        }
    } else if (lane == 16) {
        #pragma unroll
        for (int j = 0; j < 8; ++j) {
            int r = base + 8 + j;
            if (r < n_nodes) gate[r] = c[j] + bias;
        }
    }
}

// ---- per-graph max of gate ----
__global__ void segmax_kernel(const float* __restrict__ gate, const int* __restrict__ gid,
                              unsigned* __restrict__ gmax_enc, int n) {
    int i = blockIdx.x * blockDim.x + threadIdx.x;
    if (i < n) atomicMax(&gmax_enc[gid[i]], flip_f32(gate[i]));
}

// ---- e = exp(gate - gmax); ssum[g] += e; store e back into gate buffer ----
__global__ void expsum_kernel(float* __restrict__ gate, const int* __restrict__ gid,
                              const unsigned* __restrict__ gmax_enc,
                              float* __restrict__ ssum, int n) {
    int i = blockIdx.x * blockDim.x + threadIdx.x;
    if (i < n) {
        int g = gid[i];
        float e = expf(gate[i] - unflip_f32(gmax_enc[g]));
        gate[i] = e;
        atomicAdd(&ssum[g], e);
    }
}

// ---- gnorm = e / ssum[g] (in place in gate buffer) ----
__global__ void gnorm_kernel(float* __restrict__ gate, const int* __restrict__ gid,
                             const float* __restrict__ ssum, int n) {
    int i = blockIdx.x * blockDim.x + threadIdx.x;
    if (i < n) gate[i] = gate[i] / ssum[gid[i]];
}

// ---- out_i = ALPHA * feat0 (accumulator init), float4-vectorized ----
__global__ void initout_kernel(const float* __restrict__ feat0, float* __restrict__ out_i,
                               int n_vec4) {
    int i = blockIdx.x * blockDim.x + threadIdx.x;
    if (i < n_vec4) {
        float4 v = ((const float4*)feat0)[i];
        v.x *= ALPHA_F; v.y *= ALPHA_F; v.z *= ALPHA_F; v.w *= ALPHA_F;
        ((float4*)out_i)[i] = v;
    }
}

// ---- edge scatter: out_i[dst] += (1-ALPHA) * feat[src] * gnorm[src] ----
// 32 threads per edge, 4 floats (float4 gather) per thread, f32 atomic adds.
// feat table + out slice both fit in the 192MB L2 -> atomics are L2-resident.
__global__ __launch_bounds__(256) void edge_kernel(
    const float* __restrict__ feat_cur, const float* __restrict__ gnorm,
    const int* __restrict__ src, const int* __restrict__ dst,
    float* __restrict__ out_i, long long total)
{
    long long i = (long long)blockIdx.x * blockDim.x + threadIdx.x;
    if (i >= total) return;
    int e = (int)(i >> 5);
    int q = (int)(i & 31);                 // which float4 of the 128-f row
    int s = src[e];
    int d = dst[e];
    float g = (1.0f - ALPHA_F) * gnorm[s];
    float4 v = ((const float4*)(feat_cur + (size_t)s * NF))[q];
    float* o = out_i + (size_t)d * NF + q * 4;
    atomicAdd(o + 0, v.x * g);
    atomicAdd(o + 1, v.y * g);
    atomicAdd(o + 2, v.z * g);
    atomicAdd(o + 3, v.w * g);
}

extern "C" void kernel_launch(void* const* d_in, const int* in_sizes, int n_in,
                              void* d_out, int out_size, void* d_ws, size_t ws_size,
                              hipStream_t stream) {
    const float* feat = (const float*)d_in[0];   // [N,128]
    const float* W    = (const float*)d_in[1];   // [K,128]
    const float* b    = (const float*)d_in[2];   // [K]
    const int*   src  = (const int*)d_in[3];     // [E]
    const int*   dst  = (const int*)d_in[4];     // [E]
    const int*   gid  = (const int*)d_in[5];     // [N]

    const int n = in_sizes[0] / NF;
    const int e = in_sizes[3];
    const int K = in_sizes[2];
    float* out = (float*)d_out;                  // [K,N,128]

    // workspace layout (~n*4 + 512 bytes)
    float*    gate     = (float*)d_ws;                     // [n] gate/e/gnorm
    unsigned* gmax_enc = (unsigned*)(gate + n);            // [64]
    float*    ssum     = (float*)(gmax_enc + NGRAPHS);     // [64]

    const int       thr    = 256;
    const int       gN     = (n + thr - 1) / thr;
    const int       gTile  = (n + 15) / 16;
    const int       nVec4  = n * (NF / 4);
    const int       gInit  = (nVec4 + thr - 1) / thr;
    const long long totalE = (long long)e * 32;
    const unsigned  gEdge  = (unsigned)((totalE + thr - 1) / thr);

    for (int it = 0; it < K; ++it) {
        const float* fcur = (it == 0) ? feat : out + (size_t)(it - 1) * n * NF;
        float*       oi   = out + (size_t)it * n * NF;

        init_seg_kernel<<<1, 64, 0, stream>>>(gmax_enc, ssum);
        gate_wmma_kernel<<<gTile, 32, 0, stream>>>(fcur, W, b, it, gate, n);
        segmax_kernel<<<gN, thr, 0, stream>>>(gate, gid, gmax_enc, n);
        expsum_kernel<<<gN, thr, 0, stream>>>(gate, gid, gmax_enc, ssum, n);
        gnorm_kernel<<<gN, thr, 0, stream>>>(gate, gid, ssum, n);
        initout_kernel<<<gInit, thr, 0, stream>>>(feat, oi, nVec4);
        edge_kernel<<<gEdge, thr, 0, stream>>>(fcur, gate, src, dst, oi, totalE);
    }
}